// TTLayer_12584254177851
// MI455X (gfx1250) — compile-verified
//
#include <hip/hip_runtime.h>

typedef __attribute__((ext_vector_type(2))) float v2f;
typedef __attribute__((ext_vector_type(8))) float v8f;

// ---------------------------------------------------------------------------
// Kernel 1: fold the TT cores into a single 8x8 matrix  W = C3 @ C2 @ C1 @ C0
// Core layout in cores_arr (row-major views, flat offsets):
//   C0 at    0 : (24, 8)
//   C1 at  192 : (24,24)
//   C2 at  768 : (24,24)
//   C3 at 1344 : ( 8,24)
// Forward applies C3, C2, C1, C0 in that order (reversed lists), so the whole
// TT chain per 8-float row-group of x collapses to one (8x8) matrix.
// ---------------------------------------------------------------------------
__global__ void __launch_bounds__(256) tt_fold_cores(const float* __restrict__ cores,
                                                     float* __restrict__ Wout) {
    __shared__ float T1[8 * 24];
    __shared__ float T2[8 * 24];
    const float* C0 = cores + 0;     // (24, 8)
    const float* C1 = cores + 192;   // (24,24)
    const float* C2 = cores + 768;   // (24,24)
    const float* C3 = cores + 1344;  // ( 8,24)

    const int t = threadIdx.x;
    if (t < 192) {                       // T1 = C3 @ C2  : (8,24)
        int r = t / 24, c = t % 24;
        float acc = 0.f;
        for (int k = 0; k < 24; ++k) acc += C3[r * 24 + k] * C2[k * 24 + c];
        T1[t] = acc;
    }
    __syncthreads();
    if (t < 192) {                       // T2 = T1 @ C1  : (8,24)
        int r = t / 24, c = t % 24;
        float acc = 0.f;
        for (int k = 0; k < 24; ++k) acc += T1[r * 24 + k] * C1[k * 24 + c];
        T2[t] = acc;
    }
    __syncthreads();
    if (t < 64) {                        // W = T2 @ C0   : (8,8)
        int r = t / 8, c = t % 8;
        float acc = 0.f;
        for (int k = 0; k < 24; ++k) acc += T2[r * 24 + k] * C0[k * 8 + c];
        Wout[t] = acc;
    }
}

// ---------------------------------------------------------------------------
// Kernel 2: streaming GEMM  Y(M,8) = X(M,8) @ W(8,8) + bias, M = 8192*512.
//
// Work decomposition: each wave owns a fixed 16-row tile position t (0..31)
// inside the 512-row batch block and strides over batch index b. Because the
// row-within-batch is fixed, the bias fragment is loop-invariant and the
// X/Y addresses advance by a constant stride -> the inner loop is just
// 2 loads + 2 WMMAs + 8 fma-ish adds + 8 stores with immediate offsets.
//
// f32 WMMA operand layouts (wave32):
//   A 16x4  (2 VGPRs): lanes 0-15 -> K=0,1 ; lanes 16-31 -> K=2,3 (same rows)
//   B 4x16  (2 VGPRs): lanes 0-15 -> K=kb+0,kb+1 ; lanes 16-31 -> K=kb+2,kb+3
//   C/D 16x16 (8 VGPRs): vgpr v -> row v (lanes 0-15) / row v+8 (lanes 16-31),
//                        column = lane % 16
// ---------------------------------------------------------------------------
__global__ void __launch_bounds__(256) tt_gemm8x8(const float* __restrict__ X,
                                                  const float* __restrict__ Wc,   // 8x8 row-major
                                                  const float* __restrict__ bias, // 4096
                                                  float* __restrict__ Y) {
    const int lane = threadIdx.x & 31;
    const int n    = lane & 15;   // output column / A-row id within tile
    const int hi   = lane >> 4;   // 0: K pair {0,1}, 1: K pair {2,3}

    // Loop-invariant B fragments from the folded 8x8 W.
    // b0 covers K = 0..3, b1 covers K = 4..7; columns 8..15 are zero padding.
    v2f b0 = {0.f, 0.f};
    v2f b1 = {0.f, 0.f};
    if (n < 8) {
        b0.x = Wc[(0 + 2 * hi) * 8 + n];
        b0.y = Wc[(1 + 2 * hi) * 8 + n];
        b1.x = Wc[(4 + 2 * hi) * 8 + n];
        b1.y = Wc[(5 + 2 * hi) * 8 + n];
    }

    const int wave    = blockIdx.x * (blockDim.x >> 5) + (threadIdx.x >> 5);
    const int nwaves  = gridDim.x * (blockDim.x >> 5);   // 16384
    const int t       = wave & 31;                       // fixed tile-in-batch
    const int bstride = nwaves >> 5;                     // 512 batch stride
    int b             = wave >> 5;                       // starting batch

    // Loop-invariant bias fragment: row-in-batch = t*16 + v + 8*hi < 512.
    float bv[8];
#pragma unroll
    for (int v = 0; v < 8; ++v)
        bv[v] = bias[((t * 16 + v + 8 * hi) * 8 + n) & 4095];

    // Per-lane streaming pointers; advance by a constant per iteration.
    const float* xp = X + ((size_t)b * 512 + t * 16 + n) * 8 + 2 * hi;
    float*       yp = Y + ((size_t)b * 512 + t * 16 + 8 * hi) * 8 + n;
    const size_t step = (size_t)bstride * 4096;          // floats per stride

    for (; b < 8192; b += bstride, xp += step, yp += step) {
        v2f a0 = __builtin_nontemporal_load((const v2f*)xp);       // K 0..3
        v2f a1 = __builtin_nontemporal_load((const v2f*)(xp + 4)); // K 4..7

        v8f acc = {};
        acc = __builtin_amdgcn_wmma_f32_16x16x4_f32(false, a0, false, b0,
                                                    (short)0, acc, false, false);
        acc = __builtin_amdgcn_wmma_f32_16x16x4_f32(false, a1, false, b1,
                                                    (short)0, acc, false, false);

        if (n < 8) {  // the 8 real output columns; offsets are immediates
#pragma unroll
            for (int v = 0; v < 8; ++v)
                __builtin_nontemporal_store(acc[v] + bv[v], yp + v * 8);
        }
    }
}

extern "C" void kernel_launch(void* const* d_in, const int* in_sizes, int n_in,
                              void* d_out, int out_size, void* d_ws, size_t ws_size,
                              hipStream_t stream) {
    const float* x     = (const float*)d_in[0];  // (8192, 4096) fp32
    const float* cores = (const float*)d_in[1];  // 1536 fp32
    const float* bias  = (const float*)d_in[2];  // 4096 fp32
    float* out = (float*)d_out;                  // (8192, 4096) fp32
    float* Wc  = (float*)d_ws;                   // 64 fp32 scratch (folded 8x8)

    // Stage 1: fold the four TT cores into one 8x8 matrix (tiny).
    tt_fold_cores<<<1, 256, 0, stream>>>(cores, Wc);

    // Stage 2: bandwidth-bound streaming WMMA GEMM, 16384 waves,
    // each wave: fixed tile t, 16 batch iterations.
    tt_gemm8x8<<<2048, 256, 0, stream>>>(x, Wc, bias, out);
}